// DynamicRouter_39685497815918
// MI455X (gfx1250) — compile-verified
//
#include <hip/hip_runtime.h>
#include <hip/hip_bf16.h>

typedef __attribute__((ext_vector_type(2))) float v2f;
typedef __attribute__((ext_vector_type(8))) float v8f;
typedef __attribute__((ext_vector_type(4))) int   v4i;

#define TOKENS 16384
#define DM     4096
#define NE     64
#define MB     64              // rows per workgroup
#define KC     64              // K-chunk staged in LDS
#define NCHUNK (DM / KC)       // 64

// ---- gfx1250 async global->LDS path (ASYNCcnt), guarded for toolchain ----
#if defined(__has_builtin)
#  if __has_builtin(__builtin_amdgcn_global_load_async_to_lds_b128) && \
      __has_builtin(__builtin_amdgcn_s_wait_asynccnt)
#    define USE_ASYNC_LDS 1
#  endif
#endif
#ifndef USE_ASYNC_LDS
#  define USE_ASYNC_LDS 0
#endif

#if USE_ASYNC_LDS
// Builtin signature (from hipcc diagnostic): pointers to 4xi32 vectors,
// global (AS1) source, LDS (AS3) destination, imm offset, imm cpol.
typedef __attribute__((address_space(1))) v4i gv4i_t;
typedef __attribute__((address_space(3))) v4i lv4i_t;
#endif

__device__ __forceinline__ void wait_stage() {
#if USE_ASYNC_LDS
    __builtin_amdgcn_s_wait_asynccnt(0);   // s_wait_asynccnt 0
#endif
}

// Stage one K-chunk: x[64][KC] and W[KC][64] into LDS (16 B per lane per op).
__device__ __forceinline__ void stage_chunk(const float* __restrict__ x,
                                            const float* __restrict__ W,
                                            float* __restrict__ xs,   // MB*KC floats
                                            float* __restrict__ ws,   // KC*NE floats
                                            int tid, int row0, int k0)
{
#pragma unroll
    for (int i = 0; i < 4; ++i) {
        const int f = tid + i * 256;      // float4 slot 0..1023
        const int r = f >> 4;             // row 0..63
        const int c = (f & 15) << 2;      // col 0..60
        const float* xsrc = &x[(size_t)(row0 + r) * DM + k0 + c];
        const float* wsrc = &W[(size_t)(k0 + r) * NE + c];
#if USE_ASYNC_LDS
        __builtin_amdgcn_global_load_async_to_lds_b128(
            (gv4i_t*)xsrc, (lv4i_t*)&xs[r * KC + c], 0, 0);
        __builtin_amdgcn_global_load_async_to_lds_b128(
            (gv4i_t*)wsrc, (lv4i_t*)&ws[r * NE + c], 0, 0);
#else
        *reinterpret_cast<float4*>(&xs[r * KC + c]) =
            *reinterpret_cast<const float4*>(xsrc);
        *reinterpret_cast<float4*>(&ws[r * NE + c]) =
            *reinterpret_cast<const float4*>(wsrc);
#endif
    }
}

__global__ __launch_bounds__(256) void
DynamicRouter_39685497815918_kernel(const float* __restrict__ x,
                                    const float* __restrict__ W,
                                    const float* __restrict__ bvec,
                                    const float* __restrict__ gum,
                                    float* __restrict__ out)
{
    // 64 KB total: 2 x-buffers (16 KB each) + 2 W-buffers (16 KB each).
    // Logits buffer (16 KB) aliases x-buffer 0 after the GEMM finishes.
    // NOTE: buffer pointers are computed arithmetically from `smem` each use;
    // pointer arrays with LDS addrspacecast constants break ld.lld.
    __shared__ __align__(16) float smem[2 * MB * KC + 2 * KC * NE];

    const int tid   = threadIdx.x;
    const int wave  = tid >> 5;
    const int lane  = tid & 31;
    const int l16   = lane & 15;
    const int lhalf = lane >> 4;           // 0/1: half-wave selects K pair
    const int row0  = blockIdx.x * MB;

    // wave -> tile: N-tile = wave&3, M-tiles = (wave>>2)*16 and +32 (B reuse)
    const int n0  = (wave & 3) * 16;
    const int m0a = (wave >> 2) * 16;
    const int m0b = m0a + 32;

    v8f ca = {};
    v8f cb = {};

    // Prologue: async-stage chunk 0 into buffer 0.
    stage_chunk(x, W, smem, smem + 2 * MB * KC, tid, row0, 0);

    for (int i = 0; i < NCHUNK; ++i) {
        const int cur = i & 1;
        wait_stage();          // my async loads (chunk i) landed in LDS
        __syncthreads();       // all waves' data visible; prev buffer free

        if (i + 1 < NCHUNK) {  // overlap: fill other buffer while computing
            float* xn = smem + (1 - cur) * (MB * KC);
            float* wn = smem + 2 * MB * KC + (1 - cur) * (KC * NE);
            stage_chunk(x, W, xn, wn, tid, row0, (i + 1) * KC);
        }

        const float* xs = smem + cur * (MB * KC);
        const float* ws = smem + 2 * MB * KC + cur * (KC * NE);
#pragma unroll
        for (int k = 0; k < KC; k += 4) {
            const int kk = k + 2 * lhalf;
            // A frags: lane = row M, contiguous K pair -> ds_load_b64
            v2f a0 = *reinterpret_cast<const v2f*>(&xs[(m0a + l16) * KC + kk]);
            v2f a1 = *reinterpret_cast<const v2f*>(&xs[(m0b + l16) * KC + kk]);
            // B frag: lane = col N, K pair strided by row -> 2addr_stride ds load
            v2f bb;
            bb.x = ws[kk * NE + n0 + l16];
            bb.y = ws[(kk + 1) * NE + n0 + l16];
            ca = __builtin_amdgcn_wmma_f32_16x16x4_f32(false, a0, false, bb,
                                                       (short)0, ca, false, false);
            cb = __builtin_amdgcn_wmma_f32_16x16x4_f32(false, a1, false, bb,
                                                       (short)0, cb, false, false);
        }
    }

    // ---- spill C tiles to logits buffer (aliases x-buffer 0; safe: last
    // chunk computed from buffer 1, and all ds reads are barrier-ordered) ----
    float* ls = smem;          // [MB][NE]
#pragma unroll
    for (int v = 0; v < 8; ++v) {
        ls[(m0a + v + 8 * lhalf) * NE + n0 + l16] = ca[v];
        ls[(m0b + v + 8 * lhalf) * NE + n0 + l16] = cb[v];
    }
    __syncthreads();

    // ---- per-row epilogue: +bias, +gumbel, top-2, masked softmax ----
    if (tid < MB) {
        const int r = tid;
        const size_t grow = (size_t)(row0 + r) * NE;
        float best = -3.0e38f, second = -3.0e38f;
        int bi = 0, si = 0;
        for (int e = 0; e < NE; ++e) {
            float lg = ls[r * NE + e] + bvec[e];   // bias from global (L2-hot)
            ls[r * NE + e] = lg;                   // keep biased logit
            float nz = lg + gum[grow + e];
            if (nz > best)        { second = best; si = bi; best = nz; bi = e; }
            else if (nz > second) { second = nz; si = e; }
        }
        const float l0 = ls[r * NE + bi];
        const float l1 = ls[r * NE + si];
        const float mx = fmaxf(l0, l1);
        // exp(NEG_INF - mx) underflows to exact 0 in fp32 -> zeros elsewhere,
        // matching the reference masked softmax.
        const float e0 = __expf(l0 - mx);
        const float e1 = __expf(l1 - mx);
        const float inv = 1.0f / (e0 + e1);
        const float p0 = e0 * inv, p1 = e1 * inv;

        float* orow = out + grow;
        for (int e = 0; e < NE; e += 4) {
            float4 vv;
            float* pv = &vv.x;
#pragma unroll
            for (int j = 0; j < 4; ++j) {
                int ee = e + j;
                pv[j] = (ee == bi) ? p0 : ((ee == si) ? p1 : 0.0f);
            }
            *reinterpret_cast<float4*>(&orow[e]) = vv;
        }
    }
}

extern "C" void kernel_launch(void* const* d_in, const int* in_sizes, int n_in,
                              void* d_out, int out_size, void* d_ws, size_t ws_size,
                              hipStream_t stream) {
    const float* x  = (const float*)d_in[0];   // [16384, 4096]
    const float* W  = (const float*)d_in[1];   // [4096, 64]
    const float* b  = (const float*)d_in[2];   // [64]
    const float* g  = (const float*)d_in[3];   // [16384, 64]
    // d_in[4] = k (==2), baked into the top-2 epilogue
    float* out = (float*)d_out;                // [16384, 64]

    dim3 grid(TOKENS / MB);   // 256 workgroups
    dim3 block(256);          // 8 wave32 waves
    DynamicRouter_39685497815918_kernel<<<grid, block, 0, stream>>>(x, W, b, g, out);
}